// OhemCELoss2_42537356099681
// MI455X (gfx1250) — compile-verified
//
#include <hip/hip_runtime.h>
#include <hip/hip_bf16.h>

// Problem constants (fixed by the reference setup)
#define C_CLASSES 19
#define IMG_HW    (768 * 768)        // 589824
#define N_IMG     8
#define NPIX      (N_IMG * IMG_HW)   // 4,718,592  (divisible by 1024)
#define IGNORE_LB 250
// -logf(0.7f) evaluated in fp32 like the reference
#define THRESH    0.35667496f

#if __has_builtin(__builtin_amdgcn_tensor_load_to_lds) && \
    __has_builtin(__builtin_amdgcn_s_wait_tensorcnt)
#define USE_TDM 1
#else
#define USE_TDM 0
#endif

typedef unsigned int u32x4 __attribute__((ext_vector_type(4)));
typedef int          i32x4 __attribute__((ext_vector_type(4)));
typedef int          i32x8 __attribute__((ext_vector_type(8)));

// Workspace layout: [ float lossArr[NPIX] | Counters ]
struct Counters {
    unsigned int validCount;   // # pixels with target != 250
    unsigned int cntGtThresh;  // # losses > THRESH
    unsigned int kthBits;      // bit pattern of sorted[K]
    unsigned int cntGtKth;     // # losses strictly > kth
    unsigned int selBin1;      // radix select state
    unsigned int selRank1;
    unsigned int selBin2;
    unsigned int selRank2;
    double       sumGtThresh;  // f64 accumulators for fp32-accurate means
    double       sumGtKth;
    unsigned int pad[4];
    unsigned int hist1[1024];  // bits[30:21]  (sign always 0: loss >= 0)
    unsigned int hist2[2048];  // bits[20:10]
    unsigned int hist3[1024];  // bits[ 9: 0]
};

__global__ void k_init(Counters* cnt) {
    unsigned int* p = (unsigned int*)cnt;
    const int nw = (int)(sizeof(Counters) / 4);
    for (int i = threadIdx.x; i < nw; i += blockDim.x) p[i] = 0u;
}

// Main pass: fused log-softmax CE + loss store + hist1 + (>thresh) stats.
// 256 thr/block, 4 pixels/thread via float4; HW % 1024 == 0 so a block never
// straddles an image and all 19 channel loads are coalesced b128 streams.
// The block's 4KB target tile is DMA'd to LDS by the Tensor Data Mover so the
// target fetch overlaps the 19 logit-plane loads (TENSORcnt-tracked).
__global__ __launch_bounds__(256) void k_loss(const float* __restrict__ logits,
                                              const int*   __restrict__ target,
                                              float*       __restrict__ lossArr,
                                              Counters*    __restrict__ cnt) {
    __shared__ unsigned int h1[1024];
#if USE_TDM
    __shared__ __attribute__((aligned(16))) int tgt[1024];
#endif
    const int tid = threadIdx.x;

#if USE_TDM
    // Issue the TDM descriptor from wave 0 only (TDM ignores EXEC; one DMA
    // per issuing wave). 1-D tile: 1024 x 4B = this block's targets -> LDS.
    if (tid < 32) {
        const unsigned long long gaddr =
            (unsigned long long)(uintptr_t)(target + (size_t)blockIdx.x * 1024);
        // ISA 10.2: flat->LDS uses addr[31:0]; low 32 bits of a generic
        // shared-pointer are the workgroup-relative LDS byte offset.
        const unsigned lds_off = (unsigned)(uintptr_t)(const void*)&tgt[0];

        // ---- D# group 0 (128b): count | lds_addr | global_addr | type=2 ----
        u32x4 g0;
        g0.x = 1u;                                               // count=1, user mode
        g0.y = lds_off;                                          // lds_addr (bytes)
        g0.z = (unsigned)(gaddr & 0xFFFFFFFFull);                // global_addr[31:0]
        g0.w = (unsigned)((gaddr >> 32) & 0x01FFFFFFull)         // global_addr[56:32]
             | (2u << 30);                                       // type=2 ("image")

        // ---- D# group 1 (256b): data_size=4B, tensor_dim0=tile_dim0=1024 ----
        const unsigned td0 = 1024u, td1 = 1u, tile0 = 1024u;
        const unsigned long long st0 = 1024ull;                  // dim0 stride
        i32x8 g1;
        g1[0] = (int)(2u << 16);                                 // wg_mask=0, data_size=2(4B)
        g1[1] = (int)((td0 & 0xFFFFu) << 16);                    // barrier_addr=0 | td0 lo16
        g1[2] = (int)((td0 >> 16) | ((td1 & 0xFFFFu) << 16));    // td0 hi16 | td1 lo16
        g1[3] = (int)((td1 >> 16) | (tile0 << 16));              // td1 hi16 | tile_dim0
        g1[4] = 0;                                               // tile_dim1=0, tile_dim2=0 (1-D)
        g1[5] = (int)(st0 & 0xFFFFFFFFull);                      // stride0 lo32
        g1[6] = (int)((st0 >> 32) & 0xFFFFull);                  // stride0 hi16 | stride1 lo16
        g1[7] = 0;                                               // stride1 hi
        const i32x4 gz = {0, 0, 0, 0};                           // groups 2/3: <=2D tensor
#if defined(__clang_major__) && (__clang_major__ >= 23)
        const i32x8 gz8 = {0, 0, 0, 0, 0, 0, 0, 0};
        __builtin_amdgcn_tensor_load_to_lds(g0, g1, gz, gz, gz8, 0);
#else
        __builtin_amdgcn_tensor_load_to_lds(g0, g1, gz, gz, 0);
#endif
    }
#endif

    for (int i = tid; i < 1024; i += 256) h1[i] = 0u;
    __syncthreads();

    const int p0 = (blockIdx.x * 256 + tid) * 4;
    const int n  = p0 / IMG_HW;
    const int hw = p0 - n * IMG_HW;
    const float* base = logits + (size_t)n * (size_t)(C_CLASSES * IMG_HW) + hw;

    float4 x[C_CLASSES];
#pragma unroll
    for (int c = 0; c < C_CLASSES; ++c) {
        x[c] = *(const float4*)(base + (size_t)c * IMG_HW);
        // software pipeline: prefetch the 4KB chunk the next block needs from
        // this channel plane (gfx1250 global_prefetch_b8, 1 lane / 128B line)
        if ((tid & 7) == 0)
            __builtin_prefetch((const void*)(base + (size_t)c * IMG_HW + 1024), 0, 0);
    }

#if USE_TDM
    if (tid < 32) __builtin_amdgcn_s_wait_tensorcnt(0);   // DMA done (wave 0)
    __syncthreads();                                      // tgt visible to block
    const int4 t4 = *(const int4*)&tgt[tid * 4];
#else
    const int4 t4 = *(const int4*)(target + p0);
#endif
    const bool v0 = (t4.x != IGNORE_LB), v1 = (t4.y != IGNORE_LB);
    const bool v2 = (t4.z != IGNORE_LB), v3 = (t4.w != IGNORE_LB);
    const int  s0 = v0 ? t4.x : 0, s1 = v1 ? t4.y : 0;
    const int  s2 = v2 ? t4.z : 0, s3 = v3 ? t4.w : 0;

    float4 m = x[0];
#pragma unroll
    for (int c = 1; c < C_CLASSES; ++c) {
        m.x = fmaxf(m.x, x[c].x); m.y = fmaxf(m.y, x[c].y);
        m.z = fmaxf(m.z, x[c].z); m.w = fmaxf(m.w, x[c].w);
    }

    float4 e = make_float4(0.f, 0.f, 0.f, 0.f);
    float tl0 = 0.f, tl1 = 0.f, tl2 = 0.f, tl3 = 0.f;
#pragma unroll
    for (int c = 0; c < C_CLASSES; ++c) {
        e.x += __expf(x[c].x - m.x); e.y += __expf(x[c].y - m.y);
        e.z += __expf(x[c].z - m.z); e.w += __expf(x[c].w - m.w);
        if (c == s0) tl0 = x[c].x;   // v_cmp + v_cndmask (no dynamic reg indexing)
        if (c == s1) tl1 = x[c].y;
        if (c == s2) tl2 = x[c].z;
        if (c == s3) tl3 = x[c].w;
    }

    // loss = max + log(sumexp) - logit[target]; clamp tiny negatives from the
    // hw-approx exp/log so the bit pattern stays non-negative (hist bin < 1024)
    const float l0 = v0 ? fmaxf(m.x + __logf(e.x) - tl0, 0.f) : 0.f;
    const float l1 = v1 ? fmaxf(m.y + __logf(e.y) - tl1, 0.f) : 0.f;
    const float l2 = v2 ? fmaxf(m.z + __logf(e.z) - tl2, 0.f) : 0.f;
    const float l3 = v3 ? fmaxf(m.w + __logf(e.w) - tl3, 0.f) : 0.f;

    *(float4*)(lossArr + p0) = make_float4(l0, l1, l2, l3);

    atomicAdd(&h1[__float_as_uint(l0) >> 21], 1u);
    atomicAdd(&h1[__float_as_uint(l1) >> 21], 1u);
    atomicAdd(&h1[__float_as_uint(l2) >> 21], 1u);
    atomicAdd(&h1[__float_as_uint(l3) >> 21], 1u);

    int    vc = (int)v0 + (int)v1 + (int)v2 + (int)v3;
    int    gc = (l0 > THRESH) + (l1 > THRESH) + (l2 > THRESH) + (l3 > THRESH);
    double gs = (l0 > THRESH ? (double)l0 : 0.0) + (l1 > THRESH ? (double)l1 : 0.0)
              + (l2 > THRESH ? (double)l2 : 0.0) + (l3 > THRESH ? (double)l3 : 0.0);
    // wave32 reduction
    for (int o = 16; o > 0; o >>= 1) {
        vc += __shfl_down(vc, o, 32);
        gc += __shfl_down(gc, o, 32);
        gs += __shfl_down(gs, o, 32);
    }
    if ((tid & 31) == 0) {
        atomicAdd(&cnt->validCount,  (unsigned int)vc);
        atomicAdd(&cnt->cntGtThresh, (unsigned int)gc);
        atomicAdd(&cnt->sumGtThresh, gs);      // global_atomic_add_f64
    }
    __syncthreads();
    for (int i = tid; i < 1024; i += 256) {
        unsigned int v = h1[i];
        if (v) atomicAdd(&cnt->hist1[i], v);
    }
}

// --- radix select pass 2/3 histograms (loss array is L2-resident: 18.9MB < 192MB) ---
__global__ __launch_bounds__(256) void k_hist2(const float* __restrict__ lossArr,
                                               Counters*    __restrict__ cnt) {
    __shared__ unsigned int h2[2048];
    for (int i = threadIdx.x; i < 2048; i += 256) h2[i] = 0u;
    __syncthreads();
    const unsigned int bin1 = cnt->selBin1;
    const int p = (blockIdx.x * 256 + threadIdx.x) * 4;
    const float4 l = *(const float4*)(lossArr + p);
    unsigned int b;
    b = __float_as_uint(l.x); if ((b >> 21) == bin1) atomicAdd(&h2[(b >> 10) & 2047u], 1u);
    b = __float_as_uint(l.y); if ((b >> 21) == bin1) atomicAdd(&h2[(b >> 10) & 2047u], 1u);
    b = __float_as_uint(l.z); if ((b >> 21) == bin1) atomicAdd(&h2[(b >> 10) & 2047u], 1u);
    b = __float_as_uint(l.w); if ((b >> 21) == bin1) atomicAdd(&h2[(b >> 10) & 2047u], 1u);
    __syncthreads();
    for (int i = threadIdx.x; i < 2048; i += 256) {
        unsigned int v = h2[i];
        if (v) atomicAdd(&cnt->hist2[i], v);
    }
}

__global__ __launch_bounds__(256) void k_hist3(const float* __restrict__ lossArr,
                                               Counters*    __restrict__ cnt) {
    __shared__ unsigned int h3[1024];
    for (int i = threadIdx.x; i < 1024; i += 256) h3[i] = 0u;
    __syncthreads();
    const unsigned int pre = (cnt->selBin1 << 11) | cnt->selBin2;  // bits[31:10]
    const int p = (blockIdx.x * 256 + threadIdx.x) * 4;
    const float4 l = *(const float4*)(lossArr + p);
    unsigned int b;
    b = __float_as_uint(l.x); if ((b >> 10) == pre) atomicAdd(&h3[b & 1023u], 1u);
    b = __float_as_uint(l.y); if ((b >> 10) == pre) atomicAdd(&h3[b & 1023u], 1u);
    b = __float_as_uint(l.z); if ((b >> 10) == pre) atomicAdd(&h3[b & 1023u], 1u);
    b = __float_as_uint(l.w); if ((b >> 10) == pre) atomicAdd(&h3[b & 1023u], 1u);
    __syncthreads();
    for (int i = threadIdx.x; i < 1024; i += 256) {
        unsigned int v = h3[i];
        if (v) atomicAdd(&cnt->hist3[i], v);
    }
}

// --- tiny single-block selection scans (descending rank walk over LDS copy) ---
__global__ void k_select1(Counters* cnt) {
    __shared__ unsigned int h[1024];
    for (int i = threadIdx.x; i < 1024; i += blockDim.x) h[i] = cnt->hist1[i];
    __syncthreads();
    if (threadIdx.x == 0) {
        const unsigned int K = cnt->validCount >> 4;   // (#valid)//16
        unsigned int cum = 0; int b = 1023;
        for (; b > 0; --b) { unsigned int c = h[b]; if (cum + c > K) break; cum += c; }
        cnt->selBin1  = (unsigned int)b;
        cnt->selRank1 = K - cum;
    }
}

__global__ void k_select2(Counters* cnt) {
    __shared__ unsigned int h[2048];
    for (int i = threadIdx.x; i < 2048; i += blockDim.x) h[i] = cnt->hist2[i];
    __syncthreads();
    if (threadIdx.x == 0) {
        const unsigned int K = cnt->selRank1;
        unsigned int cum = 0; int b = 2047;
        for (; b > 0; --b) { unsigned int c = h[b]; if (cum + c > K) break; cum += c; }
        cnt->selBin2  = (unsigned int)b;
        cnt->selRank2 = K - cum;
    }
}

__global__ void k_select3(Counters* cnt) {
    __shared__ unsigned int h[1024];
    for (int i = threadIdx.x; i < 1024; i += blockDim.x) h[i] = cnt->hist3[i];
    __syncthreads();
    if (threadIdx.x == 0) {
        const unsigned int K = cnt->selRank2;
        unsigned int cum = 0; int b = 1023;
        for (; b > 0; --b) { unsigned int c = h[b]; if (cum + c > K) break; cum += c; }
        cnt->kthBits = (cnt->selBin1 << 21) | (cnt->selBin2 << 10) | (unsigned int)b;
    }
}

// count/sum of losses strictly greater than kth (for the top-K branch)
__global__ __launch_bounds__(256) void k_count(const float* __restrict__ lossArr,
                                               Counters*    __restrict__ cnt) {
    const float kth = __uint_as_float(cnt->kthBits);
    const int p = (blockIdx.x * 256 + threadIdx.x) * 4;
    const float4 l = *(const float4*)(lossArr + p);
    int    c = (l.x > kth) + (l.y > kth) + (l.z > kth) + (l.w > kth);
    double s = (l.x > kth ? (double)l.x : 0.0) + (l.y > kth ? (double)l.y : 0.0)
             + (l.z > kth ? (double)l.z : 0.0) + (l.w > kth ? (double)l.w : 0.0);
    for (int o = 16; o > 0; o >>= 1) {
        c += __shfl_down(c, o, 32);
        s += __shfl_down(s, o, 32);
    }
    if ((threadIdx.x & 31) == 0) {
        atomicAdd(&cnt->cntGtKth, (unsigned int)c);
        atomicAdd(&cnt->sumGtKth, s);
    }
}

__global__ void k_finalize(const Counters* __restrict__ cnt, float* __restrict__ out) {
    if (threadIdx.x == 0 && blockIdx.x == 0) {
        const float kth = __uint_as_float(cnt->kthBits);
        const unsigned int K = cnt->validCount >> 4;
        double res;
        if (kth > THRESH) {
            const unsigned int c = cnt->cntGtThresh;
            res = cnt->sumGtThresh / (c > 0 ? (double)c : 1.0);
        } else {
            const unsigned int cg = cnt->cntGtKth;   // <= K by construction
            const double s = cnt->sumGtKth + (double)(K - cg) * (double)kth;  // ties at kth
            res = s / (K > 0 ? (double)K : 1.0);
        }
        out[0] = (float)res;
    }
}

extern "C" void kernel_launch(void* const* d_in, const int* in_sizes, int n_in,
                              void* d_out, int out_size, void* d_ws, size_t ws_size,
                              hipStream_t stream) {
    (void)in_sizes; (void)n_in; (void)out_size; (void)ws_size;
    const float* logits = (const float*)d_in[0];
    const int*   target = (const int*)d_in[1];   // JAX default x64-off -> int32
    float* lossArr = (float*)d_ws;                                // 18.87 MB
    Counters* cnt  = (Counters*)((char*)d_ws + (size_t)NPIX * 4); // 256B-aligned
    float* out = (float*)d_out;

    const int blocks = NPIX / 1024;   // 4608; each block = 1024 pixels

    k_init    <<<1, 1024, 0, stream>>>(cnt);
    k_loss    <<<blocks, 256, 0, stream>>>(logits, target, lossArr, cnt);
    k_select1 <<<1, 1024, 0, stream>>>(cnt);
    k_hist2   <<<blocks, 256, 0, stream>>>(lossArr, cnt);
    k_select2 <<<1, 1024, 0, stream>>>(cnt);
    k_hist3   <<<blocks, 256, 0, stream>>>(lossArr, cnt);
    k_select3 <<<1, 1024, 0, stream>>>(cnt);
    k_count   <<<blocks, 256, 0, stream>>>(lossArr, cnt);
    k_finalize<<<1, 64, 0, stream>>>(cnt, out);
}